// SGCNet2_22694607192488
// MI455X (gfx1250) — compile-verified
//
#include <hip/hip_runtime.h>
#include <math.h>

typedef float v2f __attribute__((ext_vector_type(2)));
typedef float v8f __attribute__((ext_vector_type(8)));

#define IN_C 256
#define NHID 128
#define OUT_C 40

__device__ __forceinline__ v8f wmma_f32_16x16x4(v2f a, v2f b, v8f c) {
  // (neg_a, A, neg_b, B, c_mod, C, reuse_a, reuse_b) -> v_wmma_f32_16x16x4_f32
  return __builtin_amdgcn_wmma_f32_16x16x4_f32(false, a, false, b, (short)0, c, false, false);
}

// ---------------- graph preprocessing ----------------

__global__ void k_init(float* deg, int* cnt, int n) {
  int i = blockIdx.x * 256 + threadIdx.x;
  if (i < n) { deg[i] = 1.0f; cnt[i] = 0; }   // self-loop weight 1.0 pre-added
}

__global__ void k_count_deg(const long long* __restrict__ dst,
                            const float* __restrict__ ew,
                            int* cnt, float* deg, int e) {
  int i = blockIdx.x * 256 + threadIdx.x;
  if (i < e) {
    int d = (int)dst[i];
    atomicAdd(&cnt[d], 1);
    atomicAdd(&deg[d], ew[i]);
  }
}

__global__ void k_dinv(const float* __restrict__ deg, float* dinv, float* selfw, int n) {
  int i = blockIdx.x * 256 + threadIdx.x;
  if (i < n) {
    float d = deg[i];
    float r = (d > 0.0f) ? rsqrtf(d) : 0.0f;
    dinv[i] = r;
    selfw[i] = r * r;          // norm of the self loop: dinv*1.0*dinv
  }
}

// exclusive scan, stage 1: per-256-block scan; rp gets block-local exclusive
__global__ void k_scan1(const int* __restrict__ cnt, int* rp, int* bsum, int n) {
  __shared__ int sh[256];
  int t = threadIdx.x;
  int i = blockIdx.x * 256 + t;
  int v = (i < n) ? cnt[i] : 0;
  sh[t] = v;
  __syncthreads();
  int x = v;
  for (int off = 1; off < 256; off <<= 1) {
    int y = (t >= off) ? sh[t - off] : 0;
    __syncthreads();
    x += y;
    sh[t] = x;
    __syncthreads();
  }
  if (i < n) rp[i] = x - v;          // exclusive
  if (t == 255) bsum[blockIdx.x] = x; // block total
}

// stage 2: single-block exclusive scan of block sums (nb <= 512)
__global__ void k_scan2(int* bsum, int nb) {
  __shared__ int sh[512];
  int t = threadIdx.x;
  int v = (t < nb) ? bsum[t] : 0;
  sh[t] = v;
  __syncthreads();
  int x = v;
  for (int off = 1; off < 512; off <<= 1) {
    int y = (t >= off) ? sh[t - off] : 0;
    __syncthreads();
    x += y;
    sh[t] = x;
    __syncthreads();
  }
  if (t < nb) bsum[t] = x - v;       // exclusive
}

// stage 3: add block offsets, copy to cursor, write rp[n]=E
__global__ void k_scan3(int* rp, const int* __restrict__ bsum, int* cursor, int n, int e) {
  int i = blockIdx.x * 256 + threadIdx.x;
  if (i < n) {
    int r = rp[i] + bsum[blockIdx.x];
    rp[i] = r;
    cursor[i] = r;
  }
  if (blockIdx.x == 0 && threadIdx.x == 0) rp[n] = e;
}

// scatter edges into CSR sorted by dst; fuse gcn-norm weight computation
__global__ void k_scatter(const long long* __restrict__ src,
                          const long long* __restrict__ dst,
                          const float* __restrict__ ew,
                          const float* __restrict__ dinv,
                          int* cursor, int* csr_src, float* csr_w, int e) {
  int i = blockIdx.x * 256 + threadIdx.x;
  if (i < e) {
    int s = (int)src[i];
    int d = (int)dst[i];
    float w = dinv[s] * ew[i] * dinv[d];
    int pos = atomicAdd(&cursor[d], 1);
    csr_src[pos] = s;
    csr_w[pos] = w;
  }
}

// ---------------- GEMM 1: H[N,128] = X[N,256] @ W1[256,128] (f32 WMMA) ----------------
// block = 256 threads = 8 waves; each block owns a 16-row slab of X in LDS;
// wave w computes the 16x16 output tile at columns [16w, 16w+16).
__global__ __launch_bounds__(256) void k_gemm1(const float* __restrict__ X,
                                               const float* __restrict__ W1,
                                               float* __restrict__ H, int n) {
  __shared__ float As[16][IN_C + 1];   // stride 257 -> conflict-free column reads
  int m0 = blockIdx.x * 16;
  int tid = threadIdx.x;
  for (int i = tid; i < 16 * IN_C; i += 256) {
    int r = i >> 8, c = i & (IN_C - 1);
    int row = m0 + r; if (row >= n) row = n - 1;
    As[r][c] = X[row * IN_C + c];
  }
  __syncthreads();

  int wave = tid >> 5, lane = tid & 31;
  int n0 = wave * 16;
  int lrow = lane & 15;                 // M index for A rows / N index for B cols
  int khalf = (lane >> 4) << 1;         // lanes 0-15 -> K {0,1}; lanes 16-31 -> K {2,3}
  v8f acc = {};
#pragma unroll 4
  for (int k = 0; k < IN_C; k += 4) {
    v2f a, b;
    a.x = As[lrow][k + khalf];
    a.y = As[lrow][k + khalf + 1];
    b.x = W1[(k + khalf) * NHID + n0 + lrow];
    b.y = W1[(k + khalf + 1) * NHID + n0 + lrow];
    acc = wmma_f32_16x16x4(a, b, acc);
  }
  // C/D layout: VGPR r -> M = r (lanes 0-15) / r+8 (lanes 16-31); N = lane%16
  int rbase = m0 + ((lane >> 4) << 3);
  int col = n0 + lrow;
  if (m0 + 16 <= n) {                   // fast path: no bounds checks (N%16==0 case)
#pragma unroll
    for (int r = 0; r < 8; ++r) H[(rbase + r) * NHID + col] = acc[r];
  } else {
#pragma unroll
    for (int r = 0; r < 8; ++r) {
      int row = rbase + r;
      if (row < n) H[row * NHID + col] = acc[r];
    }
  }
}

// ---------------- propagation at C=128: out = S * in ----------------
// one wave32 per node; each lane owns 4 channels (float4) -> one b128 gather per edge per wave;
// readfirstlane forces rp/csr loads onto the scalar path (s_load).
__global__ __launch_bounds__(256) void k_prop128(const float4* __restrict__ in,
                                                 float4* __restrict__ out,
                                                 const int* __restrict__ rp,
                                                 const int* __restrict__ csr_src,
                                                 const float* __restrict__ csr_w,
                                                 const float* __restrict__ selfw, int n) {
  int wave = threadIdx.x >> 5, lane = threadIdx.x & 31;
  int node = blockIdx.x * 8 + wave;
  if (node >= n) return;
  node = __builtin_amdgcn_readfirstlane(node);

  float sw = selfw[node];
  float4 v = in[node * 32 + lane];
  float4 acc;
  acc.x = sw * v.x; acc.y = sw * v.y; acc.z = sw * v.z; acc.w = sw * v.w;

  int e = rp[node], end = rp[node + 1];
  for (; e + 4 <= end; e += 4) {
    int s0 = csr_src[e], s1 = csr_src[e + 1], s2 = csr_src[e + 2], s3 = csr_src[e + 3];
    float w0 = csr_w[e], w1 = csr_w[e + 1], w2 = csr_w[e + 2], w3 = csr_w[e + 3];
    float4 x0 = in[s0 * 32 + lane], x1 = in[s1 * 32 + lane];
    float4 x2 = in[s2 * 32 + lane], x3 = in[s3 * 32 + lane];
    acc.x = fmaf(w0, x0.x, acc.x); acc.y = fmaf(w0, x0.y, acc.y);
    acc.z = fmaf(w0, x0.z, acc.z); acc.w = fmaf(w0, x0.w, acc.w);
    acc.x = fmaf(w1, x1.x, acc.x); acc.y = fmaf(w1, x1.y, acc.y);
    acc.z = fmaf(w1, x1.z, acc.z); acc.w = fmaf(w1, x1.w, acc.w);
    acc.x = fmaf(w2, x2.x, acc.x); acc.y = fmaf(w2, x2.y, acc.y);
    acc.z = fmaf(w2, x2.z, acc.z); acc.w = fmaf(w2, x2.w, acc.w);
    acc.x = fmaf(w3, x3.x, acc.x); acc.y = fmaf(w3, x3.y, acc.y);
    acc.z = fmaf(w3, x3.z, acc.z); acc.w = fmaf(w3, x3.w, acc.w);
  }
  for (; e < end; ++e) {
    int s = csr_src[e];
    float w = csr_w[e];
    float4 xv = in[s * 32 + lane];
    acc.x = fmaf(w, xv.x, acc.x); acc.y = fmaf(w, xv.y, acc.y);
    acc.z = fmaf(w, xv.z, acc.z); acc.w = fmaf(w, xv.w, acc.w);
  }
  out[node * 32 + lane] = acc;
}

// ---------------- GEMM 2: P[N,40] = relu(H + b1) @ W2[128,40] (f32 WMMA) ----------------
// block = 128 threads = 4 waves; waves 0..2 cover the 48-wide (padded) output; bias+relu fused into A load.
__global__ __launch_bounds__(128) void k_gemm2(const float* __restrict__ Hin,
                                               const float* __restrict__ b1,
                                               const float* __restrict__ W2,
                                               float* __restrict__ P, int n) {
  __shared__ float As[16][NHID + 1];   // stride 129 -> conflict-free
  int m0 = blockIdx.x * 16;
  int tid = threadIdx.x;
  for (int i = tid; i < 16 * NHID; i += 128) {
    int r = i >> 7, c = i & (NHID - 1);
    int row = m0 + r; if (row >= n) row = n - 1;
    As[r][c] = fmaxf(Hin[row * NHID + c] + b1[c], 0.0f);
  }
  __syncthreads();

  int wave = tid >> 5, lane = tid & 31;
  if (wave < 3) {
    int n0 = wave * 16;
    int lrow = lane & 15;
    int khalf = (lane >> 4) << 1;
    int col = n0 + lrow;
    int colc = (col < OUT_C) ? col : (OUT_C - 1);   // clamp: branch-free, in-bounds
    float mask = (col < OUT_C) ? 1.0f : 0.0f;
    v8f acc = {};
#pragma unroll 4
    for (int k = 0; k < NHID; k += 4) {
      v2f a, b;
      a.x = As[lrow][k + khalf];
      a.y = As[lrow][k + khalf + 1];
      b.x = W2[(k + khalf) * OUT_C + colc] * mask;
      b.y = W2[(k + khalf + 1) * OUT_C + colc] * mask;
      acc = wmma_f32_16x16x4(a, b, acc);
    }
    int rbase = m0 + ((lane >> 4) << 3);
    if (col < OUT_C) {
      if (m0 + 16 <= n) {
#pragma unroll
        for (int r = 0; r < 8; ++r) P[(rbase + r) * OUT_C + col] = acc[r];
      } else {
#pragma unroll
        for (int r = 0; r < 8; ++r) {
          int row = rbase + r;
          if (row < n) P[row * OUT_C + col] = acc[r];
        }
      }
    }
  }
}

// ---------------- propagation at C=40 ----------------
// one wave32 per node; lanes 0..19 own 2 channels each (float2); clamped lanes keep loads in-bounds.
__global__ __launch_bounds__(256) void k_prop40(const float2* __restrict__ in,
                                                float2* __restrict__ out,
                                                const int* __restrict__ rp,
                                                const int* __restrict__ csr_src,
                                                const float* __restrict__ csr_w,
                                                const float* __restrict__ selfw, int n) {
  int wave = threadIdx.x >> 5, lane = threadIdx.x & 31;
  int node = blockIdx.x * 8 + wave;
  if (node >= n) return;
  node = __builtin_amdgcn_readfirstlane(node);
  int cl = (lane < 20) ? lane : 19;       // clamp inactive lanes to stay in-bounds

  float sw = selfw[node];
  float2 v = in[node * 20 + cl];
  float2 acc;
  acc.x = sw * v.x; acc.y = sw * v.y;

  int e = rp[node], end = rp[node + 1];
  for (; e + 4 <= end; e += 4) {
    int s0 = csr_src[e], s1 = csr_src[e + 1], s2 = csr_src[e + 2], s3 = csr_src[e + 3];
    float w0 = csr_w[e], w1 = csr_w[e + 1], w2 = csr_w[e + 2], w3 = csr_w[e + 3];
    float2 x0 = in[s0 * 20 + cl], x1 = in[s1 * 20 + cl];
    float2 x2 = in[s2 * 20 + cl], x3 = in[s3 * 20 + cl];
    acc.x = fmaf(w0, x0.x, acc.x); acc.y = fmaf(w0, x0.y, acc.y);
    acc.x = fmaf(w1, x1.x, acc.x); acc.y = fmaf(w1, x1.y, acc.y);
    acc.x = fmaf(w2, x2.x, acc.x); acc.y = fmaf(w2, x2.y, acc.y);
    acc.x = fmaf(w3, x3.x, acc.x); acc.y = fmaf(w3, x3.y, acc.y);
  }
  for (; e < end; ++e) {
    int s = csr_src[e];
    float w = csr_w[e];
    float2 xv = in[s * 20 + cl];
    acc.x = fmaf(w, xv.x, acc.x); acc.y = fmaf(w, xv.y, acc.y);
  }
  if (lane < 20) out[node * 20 + lane] = acc;
}

// ---------------- +b2 and log_softmax over 40 classes; one wave32 per node ----------------
__global__ __launch_bounds__(256) void k_logsm(float* __restrict__ out,
                                               const float* __restrict__ b2, int n) {
  int node = blockIdx.x * 8 + (threadIdx.x >> 5);
  int lane = threadIdx.x & 31;
  if (node >= n) return;
  float a = out[node * OUT_C + lane] + b2[lane];                      // c = lane (0..31)
  float bv = (lane < 8) ? (out[node * OUT_C + 32 + lane] + b2[32 + lane])
                        : -INFINITY;                                  // c = lane+32
  float m = fmaxf(a, bv);
#pragma unroll
  for (int off = 16; off >= 1; off >>= 1) m = fmaxf(m, __shfl_xor(m, off, 32));
  float s = __expf(a - m) + ((lane < 8) ? __expf(bv - m) : 0.0f);
#pragma unroll
  for (int off = 16; off >= 1; off >>= 1) s += __shfl_xor(s, off, 32);
  float lse = m + __logf(s);
  out[node * OUT_C + lane] = a - lse;
  if (lane < 8) out[node * OUT_C + 32 + lane] = bv - lse;
}

// ---------------- launch ----------------

static inline size_t alup(size_t x) { return (x + 255) & ~(size_t)255; }

extern "C" void kernel_launch(void* const* d_in, const int* in_sizes, int n_in,
                              void* d_out, int out_size, void* d_ws, size_t ws_size,
                              hipStream_t stream) {
  const float*     x   = (const float*)d_in[0];
  const long long* ei  = (const long long*)d_in[1];   // (2,E) int64: [0:E)=src, [E:2E)=dst
  const float*     ew  = (const float*)d_in[2];
  const float*     W1  = (const float*)d_in[3];
  const float*     b1  = (const float*)d_in[4];
  const float*     W2  = (const float*)d_in[5];
  const float*     b2  = (const float*)d_in[6];

  const int N = in_sizes[0] / IN_C;
  const int E = in_sizes[2];
  const long long* esrc = ei;
  const long long* edst = ei + E;

  // workspace carve
  char* w = (char*)d_ws;
  float* deg     = (float*)w; w += alup((size_t)N * 4);
  float* dinv    = (float*)w; w += alup((size_t)N * 4);
  float* selfw   = (float*)w; w += alup((size_t)N * 4);
  int*   cnt     = (int*)w;   w += alup((size_t)N * 4);      // reused as scatter cursor
  int*   rp      = (int*)w;   w += alup((size_t)(N + 1) * 4);
  int*   bsum    = (int*)w;   w += alup((size_t)512 * 4);
  int*   csr_src = (int*)w;   w += alup((size_t)E * 4);
  float* csr_w   = (float*)w; w += alup((size_t)E * 4);
  float* Hbuf    = (float*)w; w += alup((size_t)N * NHID * 4);
  float* Tbuf    = (float*)w; w += alup((size_t)N * NHID * 4);
  float* P1      = (float*)w; w += alup((size_t)N * OUT_C * 4);
  float* P2      = (float*)w; w += alup((size_t)N * OUT_C * 4);

  const int nbN = (N + 255) / 256;           // 391 for N=100000 (<=512, fits scan2)
  const int nbE = (E + 255) / 256;
  const int mt  = (N + 15) / 16;             // 6250 GEMM row tiles
  const int nw  = (N + 7) / 8;               // 8 nodes (waves) per block

  // graph normalization + CSR build (counting sort by dst)
  k_init     <<<nbN, 256, 0, stream>>>(deg, cnt, N);
  k_count_deg<<<nbE, 256, 0, stream>>>(edst, ew, cnt, deg, E);
  k_dinv     <<<nbN, 256, 0, stream>>>(deg, dinv, selfw, N);
  k_scan1    <<<nbN, 256, 0, stream>>>(cnt, rp, bsum, N);
  k_scan2    <<<1,   512, 0, stream>>>(bsum, nbN);
  k_scan3    <<<nbN, 256, 0, stream>>>(rp, bsum, cnt, N, E);
  k_scatter  <<<nbE, 256, 0, stream>>>(esrc, edst, ew, dinv, cnt, csr_src, csr_w, E);

  // layer 1: H = X@W1, then S^2 (bias+relu deferred into GEMM2's A-load)
  k_gemm1  <<<mt, 256, 0, stream>>>(x, W1, Hbuf, N);
  k_prop128<<<nw, 256, 0, stream>>>((const float4*)Hbuf, (float4*)Tbuf, rp, csr_src, csr_w, selfw, N);
  k_prop128<<<nw, 256, 0, stream>>>((const float4*)Tbuf, (float4*)Hbuf, rp, csr_src, csr_w, selfw, N);

  // layer 2: P = relu(H+b1)@W2, then S^2 into d_out, then +b2 / log_softmax
  k_gemm2 <<<mt, 128, 0, stream>>>(Hbuf, b1, W2, P1, N);
  k_prop40<<<nw, 256, 0, stream>>>((const float2*)P1, (float2*)P2, rp, csr_src, csr_w, selfw, N);
  k_prop40<<<nw, 256, 0, stream>>>((const float2*)P2, (float2*)d_out, rp, csr_src, csr_w, selfw, N);
  k_logsm <<<(N + 7) / 8, 256, 0, stream>>>((float*)d_out, b2, N);
}